// GraphAttentionLayer_23691039604829
// MI455X (gfx1250) — compile-verified
//
#include <hip/hip_runtime.h>
#include <hip/hip_bf16.h>
#include <stdint.h>

#define DIM 64
#define HEADS 4
#define CH 16
#define NEG_SLOPE 0.2f
#define LN_EPS 1e-5f

typedef float v2f __attribute__((ext_vector_type(2)));
typedef float v8f __attribute__((ext_vector_type(8)));

// ---------------------------------------------------------------------------
// 0) init: zero the output accumulator, set segment-max to -inf, seg-sum to 0
// ---------------------------------------------------------------------------
__global__ __launch_bounds__(256) void gat_init_kernel(float* __restrict__ out,
                                                       float* __restrict__ maxv,
                                                       float* __restrict__ sumv,
                                                       int n) {
    int i = blockIdx.x * blockDim.x + threadIdx.x;
    int stride = gridDim.x * blockDim.x;
    for (int j = i; j < n * DIM; j += stride) out[j] = 0.0f;
    for (int j = i; j < n * HEADS; j += stride) {
        maxv[j] = __int_as_float(0xFF800000);  // -inf (bits valid for int/uint atomic trick)
        sumv[j] = 0.0f;
    }
}

// ---------------------------------------------------------------------------
// 1) h = x @ W via V_WMMA_F32_16X16X4_F32.
//    One wave computes one 16x16 output tile (16 rows x 1 head).
//    4 waves per block -> a 16-row slab of all 64 output columns.
// ---------------------------------------------------------------------------
__global__ __launch_bounds__(128) void gat_gemm_wmma_kernel(const float* __restrict__ x,
                                                            const float* __restrict__ W,
                                                            float* __restrict__ h,
                                                            int n) {
    const int row0 = blockIdx.x * 16;
    const int wave = threadIdx.x >> 5;      // 0..3 -> 16-col tile (== head)
    const int lane = threadIdx.x & 31;
    const int mn   = lane & 15;             // M for A-load, N for B-load
    const int khalf = (lane >> 4) << 1;     // lanes 0-15 -> K{0,1}, lanes 16-31 -> K{2,3}
    const int col0 = wave * 16;

    int arow = row0 + mn;
    if (arow >= n) arow = n - 1;            // clamp (keeps EXEC all-ones for WMMA)

    v8f c = {};
    #pragma unroll
    for (int kb = 0; kb < DIM / 4; ++kb) {
        const int k = kb * 4 + khalf;
        v2f a, b;
        // A 16x4 (MxK): lane holds (M=mn, K=k..k+1)
        a.x = x[arow * DIM + k];
        a.y = x[arow * DIM + k + 1];
        // B 4x16 (KxN): lane holds (K=k..k+1, N=mn)
        b.x = W[(k    ) * DIM + col0 + mn];
        b.y = W[(k + 1) * DIM + col0 + mn];
        c = __builtin_amdgcn_wmma_f32_16x16x4_f32(false, a, false, b,
                                                  (short)0, c, false, false);
    }

    // C/D layout: VGPR v -> M = v + 8*(lane>=16), N = lane&15
    const int ncol  = col0 + mn;
    const int mbase = (lane >> 4) * 8;
    #pragma unroll
    for (int v = 0; v < 8; ++v) {
        const int gr = row0 + mbase + v;
        if (gr < n) h[gr * DIM + ncol] = c[v];
    }
}

// ---------------------------------------------------------------------------
// 2) per-(node,head) attention logits: a_src/a_dst = <h[n,head,:], att[head,:]>
// ---------------------------------------------------------------------------
__global__ __launch_bounds__(256) void gat_node_att_kernel(const float* __restrict__ h,
                                                           const float* __restrict__ att_src,
                                                           const float* __restrict__ att_dst,
                                                           float* __restrict__ a_src,
                                                           float* __restrict__ a_dst,
                                                           int n) {
    int t = blockIdx.x * blockDim.x + threadIdx.x;
    if (t >= n * HEADS) return;
    const int node = t >> 2;
    const int head = t & 3;
    const float* hp = h + node * DIM + head * CH;
    float s = 0.0f, d = 0.0f;
    #pragma unroll
    for (int c = 0; c < CH; ++c) {
        const float v = hp[c];
        s += v * att_src[head * CH + c];
        d += v * att_dst[head * CH + c];
    }
    a_src[t] = s;
    a_dst[t] = d;
}

__device__ __forceinline__ float leaky(float e) {
    return e > 0.0f ? e : e * NEG_SLOPE;
}

// float atomic-max via native integer atomics (monotone bit encodings)
__device__ __forceinline__ void atomicMaxF32(float* addr, float val) {
    if (!(val < 0.0f)) {
        atomicMax((int*)addr, __float_as_int(val));
    } else {
        atomicMin((unsigned int*)addr, __float_as_uint(val));
    }
}

// ---------------------------------------------------------------------------
// 3) edge pass A: segment max of leaky_relu(a_src[src]+a_dst[dst]) per head
// ---------------------------------------------------------------------------
__global__ __launch_bounds__(256) void gat_edge_max_kernel(const long long* __restrict__ ei,
                                                           const float* __restrict__ a_src,
                                                           const float* __restrict__ a_dst,
                                                           float* __restrict__ maxv,
                                                           int E, int Etot) {
    int e = blockIdx.x * blockDim.x + threadIdx.x;
    if (e >= Etot) return;
    long long s, d;
    if (e < E) { s = ei[e]; d = ei[E + e]; } else { s = d = (long long)(e - E); }
    #pragma unroll
    for (int hh = 0; hh < HEADS; ++hh) {
        const float v = leaky(a_src[s * HEADS + hh] + a_dst[d * HEADS + hh]);
        atomicMaxF32(&maxv[d * HEADS + hh], v);
    }
}

// ---------------------------------------------------------------------------
// 4) edge pass B: segment sum of exp(e - max)
// ---------------------------------------------------------------------------
__global__ __launch_bounds__(256) void gat_edge_sum_kernel(const long long* __restrict__ ei,
                                                           const float* __restrict__ a_src,
                                                           const float* __restrict__ a_dst,
                                                           const float* __restrict__ maxv,
                                                           float* __restrict__ sumv,
                                                           int E, int Etot) {
    int e = blockIdx.x * blockDim.x + threadIdx.x;
    if (e >= Etot) return;
    long long s, d;
    if (e < E) { s = ei[e]; d = ei[E + e]; } else { s = d = (long long)(e - E); }
    #pragma unroll
    for (int hh = 0; hh < HEADS; ++hh) {
        const float v  = leaky(a_src[s * HEADS + hh] + a_dst[d * HEADS + hh]);
        const float ex = __expf(v - maxv[d * HEADS + hh]);
        atomicAdd(&sumv[d * HEADS + hh], ex);
    }
}

// ---------------------------------------------------------------------------
// 5) edge pass C: out[dst] += alpha * h[src]. One wave32 per edge, 2 ch/lane
//    -> coalesced 128B reads of h[src] and coalesced atomic bursts to out.
// ---------------------------------------------------------------------------
__global__ __launch_bounds__(256) void gat_edge_msg_kernel(const long long* __restrict__ ei,
                                                           const float* __restrict__ h,
                                                           const float* __restrict__ a_src,
                                                           const float* __restrict__ a_dst,
                                                           const float* __restrict__ maxv,
                                                           const float* __restrict__ sumv,
                                                           float* __restrict__ out,
                                                           int E, int Etot) {
    const int widx = (blockIdx.x * blockDim.x + threadIdx.x) >> 5;  // edge index
    const int lane = threadIdx.x & 31;
    if (widx >= Etot) return;
    long long s, d;
    if (widx < E) { s = ei[widx]; d = ei[E + widx]; } else { s = d = (long long)(widx - E); }
    #pragma unroll
    for (int half = 0; half < 2; ++half) {
        const int c = lane + half * 32;
        const int hh = c >> 4;  // head = channel / 16
        const float e  = leaky(a_src[s * HEADS + hh] + a_dst[d * HEADS + hh]);
        const float al = __expf(e - maxv[d * HEADS + hh]) / (sumv[d * HEADS + hh] + 1e-16f);
        atomicAdd(&out[d * DIM + c], h[s * DIM + c] * al);
    }
}

// ---------------------------------------------------------------------------
// 6) residual + LayerNorm: one wave32 per node row, butterfly reduction
// ---------------------------------------------------------------------------
__global__ __launch_bounds__(256) void gat_residual_ln_kernel(const float* __restrict__ x,
                                                              float* __restrict__ out,
                                                              const float* __restrict__ bias,
                                                              const float* __restrict__ gamma,
                                                              const float* __restrict__ beta,
                                                              int n) {
    const int node = (blockIdx.x * blockDim.x + threadIdx.x) >> 5;
    const int lane = threadIdx.x & 31;
    if (node >= n) return;
    const int c0 = lane;
    const int c1 = lane + 32;
    float y0 = x[node * DIM + c0] + out[node * DIM + c0] + bias[c0];
    float y1 = x[node * DIM + c1] + out[node * DIM + c1] + bias[c1];
    float sum = y0 + y1;
    float sq  = y0 * y0 + y1 * y1;
    #pragma unroll
    for (int off = 16; off > 0; off >>= 1) {
        sum += __shfl_xor(sum, off, 32);
        sq  += __shfl_xor(sq,  off, 32);
    }
    const float mu  = sum * (1.0f / DIM);
    const float var = sq * (1.0f / DIM) - mu * mu;
    const float inv = rsqrtf(var + LN_EPS);
    out[node * DIM + c0] = (y0 - mu) * inv * gamma[c0] + beta[c0];
    out[node * DIM + c1] = (y1 - mu) * inv * gamma[c1] + beta[c1];
}

// ---------------------------------------------------------------------------
extern "C" void kernel_launch(void* const* d_in, const int* in_sizes, int n_in,
                              void* d_out, int out_size, void* d_ws, size_t ws_size,
                              hipStream_t stream) {
    const float*     x       = (const float*)d_in[0];
    const long long* ei      = (const long long*)d_in[1];   // int64 edge_index [2,E]
    const float*     W       = (const float*)d_in[2];
    const float*     att_src = (const float*)d_in[3];
    const float*     att_dst = (const float*)d_in[4];
    const float*     bias    = (const float*)d_in[5];
    const float*     gamma   = (const float*)d_in[6];
    const float*     beta    = (const float*)d_in[7];

    const int n    = in_sizes[0] / DIM;
    const int E    = in_sizes[1] / 2;
    const int Etot = E + n;  // + self loops

    float* out = (float*)d_out;

    // workspace layout: h [n*64] | a_src [n*4] | a_dst [n*4] | maxv [n*4] | sumv [n*4]
    float* h     = (float*)d_ws;
    float* a_src = h + (size_t)n * DIM;
    float* a_dst = a_src + (size_t)n * HEADS;
    float* maxv  = a_dst + (size_t)n * HEADS;
    float* sumv  = maxv + (size_t)n * HEADS;

    gat_init_kernel<<<1024, 256, 0, stream>>>(out, maxv, sumv, n);

    gat_gemm_wmma_kernel<<<(n + 15) / 16, 128, 0, stream>>>(x, W, h, n);

    gat_node_att_kernel<<<(n * HEADS + 255) / 256, 256, 0, stream>>>(h, att_src, att_dst,
                                                                     a_src, a_dst, n);

    gat_edge_max_kernel<<<(Etot + 255) / 256, 256, 0, stream>>>(ei, a_src, a_dst, maxv, E, Etot);

    gat_edge_sum_kernel<<<(Etot + 255) / 256, 256, 0, stream>>>(ei, a_src, a_dst, maxv, sumv,
                                                                E, Etot);

    gat_edge_msg_kernel<<<(Etot + 7) / 8, 256, 0, stream>>>(ei, h, a_src, a_dst, maxv, sumv,
                                                            out, E, Etot);

    gat_residual_ln_kernel<<<(n + 7) / 8, 256, 0, stream>>>(x, out, bias, gamma, beta, n);
}